// ScaledDotProductAttention_25520695672864
// MI455X (gfx1250) — compile-verified
//
#include <hip/hip_runtime.h>

typedef float v2f __attribute__((ext_vector_type(2)));
typedef float v4f __attribute__((ext_vector_type(4)));
typedef float v8f __attribute__((ext_vector_type(8)));

#define B_      4
#define S_      4096
#define D_      64
#define INV_T   0.125f          // 1 / temperature (= 1/sqrt(64))
#define RSTRIDE 4100            // padded LDS row stride (floats): 4100 mod 64 = 4 -> conflict-free gathers
#define NWAVES  8
#define KTILES  (S_ / 16)       // 256

// D = A(16x4, f32) * B(4x16, f32) + C(16x16, f32)   -> v_wmma_f32_16x16x4_f32
__device__ __forceinline__ v8f wmma4(v2f a, v2f b, v8f c) {
    return __builtin_amdgcn_wmma_f32_16x16x4_f32(
        /*neg_a=*/false, a, /*neg_b=*/false, b,
        /*c_mod=*/(short)0, c, /*reuse_a=*/false, /*reuse_b=*/false);
}

__global__ void __launch_bounds__(256, 1)
attn_fused_kernel(const float* __restrict__ Q,
                  const float* __restrict__ K,
                  const float* __restrict__ V,
                  float* __restrict__ out,
                  float* __restrict__ attn)
{
    extern __shared__ float smem[];
    float* sc      = smem;                       // 16 x RSTRIDE score stripe (262.4 KB)
    float* outacc  = smem + 16 * RSTRIDE;        // 16 x 64 P*V accumulator
    float* rsuminv = outacc + 16 * D_;           // 16 reciprocal row sums

    const int tid  = threadIdx.x;
    const int lane = tid & 31;
    const int wave = tid >> 5;
    const int wg   = blockIdx.x;
    const int b    = wg >> 8;                    // wg / 256
    const int qbase = (wg & 255) << 4;           // 16-row Q tile base

    const int n   = lane & 15;                   // column / M index within half-wave
    const int rb  = (lane < 16) ? 0 : 8;         // C/D row base for this lane half
    const int csh = (lane < 16) ? 0 : 2;         // A/B K-dim shift for this lane half

    // zero the P*V accumulator
    for (int i = tid; i < 16 * D_; i += 256) outacc[i] = 0.0f;

    // ---------------- load Q A-fragments once (16 chunks of Kdim=4) ----------------
    v2f a[16];
    {
        const float* qrow = Q + ((size_t)b * S_ + qbase + n) * D_ + csh;
        #pragma unroll
        for (int c = 0; c < 16; ++c)
            a[c] = *(const v2f*)(qrow + 4 * c);
    }

    // ---------------- QK^T: each wave handles k-tiles wave, wave+8, ... ------------
    for (int jt = wave; jt < KTILES; jt += NWAVES) {
        const int kbase = jt << 4;
        const float* krow = K + ((size_t)b * S_ + kbase + n) * D_ + csh;
        v8f acc = {0.f, 0.f, 0.f, 0.f, 0.f, 0.f, 0.f, 0.f};
        #pragma unroll
        for (int c = 0; c < 16; ++c) {
            v2f bf = *(const v2f*)(krow + 4 * c);   // B[d][ncol] = K[kbase+ncol][d]
            acc = wmma4(a[c], bf, acc);
        }
        float* dst = sc + kbase + n;
        #pragma unroll
        for (int r = 0; r < 8; ++r)
            dst[(size_t)(r + rb) * RSTRIDE] = acc[r] * INV_T;
    }

    __syncthreads();

    // ---------------- softmax over each row (wave w owns rows 2w, 2w+1) ------------
    #pragma unroll
    for (int rr = 0; rr < 2; ++rr) {
        const int row = wave * 2 + rr;
        float* rp = sc + (size_t)row * RSTRIDE;
        float m = -3.402823466e38f;
        for (int i = lane * 4; i < S_; i += 128) {
            v4f v = *(v4f*)(rp + i);
            m = fmaxf(m, fmaxf(fmaxf(v.x, v.y), fmaxf(v.z, v.w)));
        }
        #pragma unroll
        for (int off = 16; off > 0; off >>= 1)
            m = fmaxf(m, __shfl_xor(m, off, 32));

        float s = 0.f;
        for (int i = lane * 4; i < S_; i += 128) {
            v4f v = *(v4f*)(rp + i);
            v.x = __expf(v.x - m); v.y = __expf(v.y - m);
            v.z = __expf(v.z - m); v.w = __expf(v.w - m);
            *(v4f*)(rp + i) = v;                 // scores -> unnormalized probs in place
            s += (v.x + v.y) + (v.z + v.w);
        }
        #pragma unroll
        for (int off = 16; off > 0; off >>= 1)
            s += __shfl_xor(s, off, 32);
        if (lane == 0) rsuminv[row] = 1.0f / s;
    }

    __syncthreads();

    // ------- stream normalized attn to global: non-temporal (write-once, 256 MB) ---
    // NT store policy keeps the 192 MB L2 reserved for the re-used K/V working set.
    {
        float* attnb = attn + ((size_t)b * S_ + qbase) * S_;
        for (int i = tid; i < 16 * (S_ / 4); i += 256) {
            const int row = i >> 10;             // 1024 float4 per row
            const int c4  = i & 1023;
            v4f p = *(v4f*)(sc + (size_t)row * RSTRIDE + c4 * 4);
            const float f = rsuminv[row];
            p.x *= f; p.y *= f; p.z *= f; p.w *= f;
            __builtin_nontemporal_store(p, (v4f*)(attnb + (size_t)row * S_ + c4 * 4));
        }
    }

    // ---------------- P * V via WMMA: waves split K=4096 into 512-wide slabs --------
    {
        v8f acc0 = {0.f,0.f,0.f,0.f,0.f,0.f,0.f,0.f};
        v8f acc1 = acc0, acc2 = acc0, acc3 = acc0;
        const int k0base = wave * (S_ / NWAVES);          // 512 per wave
        const float* vb = V + (size_t)b * S_ * D_;
        for (int kc = 0; kc < (S_ / NWAVES) / 4; ++kc) {  // 128 K-chunks of 4
            const int k0 = k0base + kc * 4 + csh;
            v2f af = *(v2f*)(sc + (size_t)n * RSTRIDE + k0);   // A[m][k] = P[n][k0..k0+1]
            const float* v0 = vb + (size_t)k0 * D_ + n;        // B[k][ncol] = V[k0..][ntile*16+n]
            v2f b0 = { v0[0],  v0[D_ + 0]  };
            acc0 = wmma4(af, b0, acc0);
            v2f b1 = { v0[16], v0[D_ + 16] };
            acc1 = wmma4(af, b1, acc1);
            v2f b2 = { v0[32], v0[D_ + 32] };
            acc2 = wmma4(af, b2, acc2);
            v2f b3 = { v0[48], v0[D_ + 48] };
            acc3 = wmma4(af, b3, acc3);
        }
        // cross-wave reduction in LDS (ds_add_f32)
        #pragma unroll
        for (int r = 0; r < 8; ++r) {
            atomicAdd(&outacc[(r + rb) * D_ +  0 + n], acc0[r]);
            atomicAdd(&outacc[(r + rb) * D_ + 16 + n], acc1[r]);
            atomicAdd(&outacc[(r + rb) * D_ + 32 + n], acc2[r]);
            atomicAdd(&outacc[(r + rb) * D_ + 48 + n], acc3[r]);
        }
    }

    __syncthreads();

    // ---------------- scale by 1/rowsum and store out (non-temporal) ----------------
    for (int t = tid * 4; t < 16 * D_; t += 256 * 4) {
        const int m = t >> 6;                    // 64 floats per row, t is 4-aligned
        v4f o = *(v4f*)(outacc + t);
        const float f = rsuminv[m];
        o.x *= f; o.y *= f; o.z *= f; o.w *= f;
        __builtin_nontemporal_store(o, (v4f*)(out + ((size_t)b * S_ + qbase + m) * D_ + (t & 63)));
    }
}

extern "C" void kernel_launch(void* const* d_in, const int* in_sizes, int n_in,
                              void* d_out, int out_size, void* d_ws, size_t ws_size,
                              hipStream_t stream) {
    const float* q = (const float*)d_in[0];
    const float* k = (const float*)d_in[1];
    const float* v = (const float*)d_in[2];
    float* out  = (float*)d_out;
    float* attn = out + (size_t)B_ * S_ * D_;   // tuple: (out, attn) concatenated flat

    const size_t ldsBytes = (size_t)(16 * RSTRIDE + 16 * D_ + 16) * sizeof(float); // ~266.6 KB < 320 KB
    hipFuncSetAttribute((const void*)attn_fused_kernel,
                        hipFuncAttributeMaxDynamicSharedMemorySize, (int)ldsBytes);

    attn_fused_kernel<<<dim3(B_ * (S_ / 16)), dim3(256), ldsBytes, stream>>>(q, k, v, out, attn);
}